// ClusterBasedSpatialAwarenessModule_43413529428136
// MI455X (gfx1250) — compile-verified
//
#include <hip/hip_runtime.h>
#include <cstddef>

#define S_DIM 1024
#define B_DIM 8
#define D_DIM 512
#define NROWS 8192          // B*S flat rows
#define EPSV  1e-12f

typedef __attribute__((ext_vector_type(2))) float v2f;
typedef __attribute__((ext_vector_type(4))) float f4;
typedef __attribute__((ext_vector_type(8))) float v8f;

__device__ __forceinline__ float wred32(float v) {
    #pragma unroll
    for (int m = 16; m >= 1; m >>= 1) v += __shfl_xor(v, m, 32);
    return v;
}
__device__ __forceinline__ float dot4(f4 a, f4 b) {
    f4 p = a * b;
    return p.x + p.y + p.z + p.w;
}

// ---------------------------------------------------------------------------
// Kernel 1: tfm[m] = X (8192x512) * W[m]^T (512x512) + b[m], f32 WMMA 16x16x4.
// Each wave: 64(M) x 16(N) tile, K stepped by 4. 12288 wave-tiles / 8 = 1536 blocks.
// A frag (16x4 f32): lane(0-15)=row M, VGPR{0,1}=K{0,1}; lanes 16-31 K{2,3}
//   -> per-lane v2f load at x[(m0+lo)*D + k0 + 2*hi]
// B frag (4x16 f32): B[k,n] = W[n,k]; same mirrored layout
//   -> per-lane v2f load at W[(n0+lo)*D + k0 + 2*hi]
// ---------------------------------------------------------------------------
__global__ __launch_bounds__(256) void gemm_wmma_f32(
    const float* __restrict__ x, const float* __restrict__ W,
    const float* __restrict__ bias, float* __restrict__ tfm)
{
    const int wid  = blockIdx.x * 8 + (threadIdx.x >> 5);
    const int lane = threadIdx.x & 31;
    const int lo   = lane & 15;
    const int hi   = lane >> 4;

    const int plane = wid >> 12;        // / 4096 tiles per plane
    const int rem   = wid & 4095;
    const int m0    = (rem >> 5) * 64;  // 128 M-supertiles of 64 rows
    const int n0    = (rem & 31) * 16;  // 32 N-tiles of 16 cols

    const float* Wm = W + (size_t)plane * D_DIM * D_DIM;
    const float* a0p = x + (size_t)(m0 +  0 + lo) * D_DIM + 2 * hi;
    const float* a1p = x + (size_t)(m0 + 16 + lo) * D_DIM + 2 * hi;
    const float* a2p = x + (size_t)(m0 + 32 + lo) * D_DIM + 2 * hi;
    const float* a3p = x + (size_t)(m0 + 48 + lo) * D_DIM + 2 * hi;
    const float* bp  = Wm + (size_t)(n0 + lo) * D_DIM + 2 * hi;

    v8f acc0 = {}, acc1 = {}, acc2 = {}, acc3 = {};

    #pragma unroll 4
    for (int k0 = 0; k0 < D_DIM; k0 += 4) {
        v2f bf = *(const v2f*)(bp  + k0);
        v2f a0 = *(const v2f*)(a0p + k0);
        v2f a1 = *(const v2f*)(a1p + k0);
        v2f a2 = *(const v2f*)(a2p + k0);
        v2f a3 = *(const v2f*)(a3p + k0);
        acc0 = __builtin_amdgcn_wmma_f32_16x16x4_f32(false, a0, false, bf, (short)0, acc0, false, false);
        acc1 = __builtin_amdgcn_wmma_f32_16x16x4_f32(false, a1, false, bf, (short)0, acc1, false, false);
        acc2 = __builtin_amdgcn_wmma_f32_16x16x4_f32(false, a2, false, bf, (short)0, acc2, false, false);
        acc3 = __builtin_amdgcn_wmma_f32_16x16x4_f32(false, a3, false, bf, (short)0, acc3, false, false);
    }

    const float bv = bias[(size_t)plane * D_DIM + n0 + lo];
    float* tp = tfm + (size_t)plane * NROWS * D_DIM;
    // C/D layout: VGPR v -> row (v + 8*hi), col = lo
    #pragma unroll
    for (int v = 0; v < 8; ++v) {
        int r = v + 8 * hi;
        tp[(size_t)(m0 +  0 + r) * D_DIM + n0 + lo] = acc0[v] + bv;
        tp[(size_t)(m0 + 16 + r) * D_DIM + n0 + lo] = acc1[v] + bv;
        tp[(size_t)(m0 + 32 + r) * D_DIM + n0 + lo] = acc2[v] + bv;
        tp[(size_t)(m0 + 48 + r) * D_DIM + n0 + lo] = acc3[v] + bv;
    }
}

// ---------------------------------------------------------------------------
// Kernel 2: windowed similarity-weighted accumulation.
// Block = (b, s0..s0+7). Stages 22 x-rows [s0-7, s0+14] into LDS once,
// computes inverse row norms, then each wave handles one s.
// ---------------------------------------------------------------------------
__global__ __launch_bounds__(256) void window_fuse(
    const float* __restrict__ x, const float* __restrict__ tfm,
    const float* __restrict__ alphas, const float* __restrict__ betas,
    float* __restrict__ out)
{
    __shared__ float lx[22 * D_DIM];
    __shared__ float linv[22];

    const int b   = blockIdx.x >> 7;          // / 128
    const int s0  = (blockIdx.x & 127) * 8;
    const int tid = threadIdx.x;

    // ---- stage 22 rows into LDS (each thread: one float2 per row) ----
    for (int t = 0; t < 22; ++t) {
        const int pos = s0 - 7 + t;
        v2f v = {0.f, 0.f};
        if (pos >= 0 && pos < S_DIM)
            v = *(const v2f*)(x + ((size_t)pos * B_DIM + b) * D_DIM + tid * 2);
        *(v2f*)&lx[t * D_DIM + tid * 2] = v;
    }
    __syncthreads();

    const int wave = tid >> 5;
    const int lane = tid & 31;
    const int off0 = 4 * lane;                // per-lane f4 offsets: off0 + 128*j

    // ---- inverse norms of the 22 rows (waves cooperate) ----
    for (int t = wave; t < 22; t += 8) {
        float p = 0.f;
        #pragma unroll
        for (int j = 0; j < 4; ++j) {
            f4 v = *(const f4*)&lx[t * D_DIM + off0 + 128 * j];
            p += dot4(v, v);
        }
        p = wred32(p);
        if (lane == 0) linv[t] = 1.0f / fmaxf(sqrtf(p), EPSV);
    }
    __syncthreads();

    const int s = s0 + wave;

    // acc starts as xb row s (LDS slot wave+7)
    f4 acc[4], csum[4], cvec[4];
    #pragma unroll
    for (int j = 0; j < 4; ++j) {
        acc[j]  = *(const f4*)&lx[(wave + 7) * D_DIM + off0 + 128 * j];
        csum[j] = (f4){0.f, 0.f, 0.f, 0.f};
    }

    const int radii[3] = {3, 5, 7};
    int cnt = 0, prevr = 0;

    for (int m = 0; m < 3; ++m) {
        const int r = radii[m];
        // extend nested window sum with newly covered offsets
        for (int w = -r; w <= r; ++w) {
            const int aw = w < 0 ? -w : w;
            if (m > 0 && aw <= prevr) continue;
            const int pos = s + w;
            if (pos < 0 || pos >= S_DIM) continue;
            const float* row = &lx[(wave + 7 + w) * D_DIM];
            #pragma unroll
            for (int j = 0; j < 4; ++j)
                csum[j] += *(const f4*)(row + off0 + 128 * j);
            ++cnt;
        }
        prevr = r;

        const float icnt = 1.0f / (float)cnt;
        float cn2 = 0.f;
        #pragma unroll
        for (int j = 0; j < 4; ++j) {
            cvec[j] = csum[j] * icnt;
            cn2 += dot4(cvec[j], cvec[j]);
        }
        cn2 = wred32(cn2);
        const float cinv = 1.0f / fmaxf(sqrtf(cn2), EPSV);

        const float alpha = alphas[m];
        const float beta  = betas[m];
        const float* tp = tfm + (size_t)m * NROWS * D_DIM;

        for (int w = -r; w <= r; ++w) {
            const int pos = s + w;
            if (pos < 0 || pos >= S_DIM) continue;
            const float* row = &lx[(wave + 7 + w) * D_DIM];
            float p = 0.f;
            #pragma unroll
            for (int j = 0; j < 4; ++j) {
                f4 xw = *(const f4*)(row + off0 + 128 * j);
                p += dot4(xw, cvec[j]);
            }
            p = wred32(p);
            const float sim = p * cinv * linv[wave + 7 + w];
            const float wt  = 1.0f / (1.0f + __expf(-(alpha * sim + beta)));
            const float* trow = tp + ((size_t)pos * B_DIM + b) * D_DIM;
            #pragma unroll
            for (int j = 0; j < 4; ++j)
                acc[j] += wt * *(const f4*)(trow + off0 + 128 * j);
        }
    }

    float* orow = out + ((size_t)s * B_DIM + b) * D_DIM;
    #pragma unroll
    for (int j = 0; j < 4; ++j)
        *(f4*)(orow + off0 + 128 * j) = acc[j];
}

// ---------------------------------------------------------------------------
extern "C" void kernel_launch(void* const* d_in, const int* in_sizes, int n_in,
                              void* d_out, int out_size, void* d_ws, size_t ws_size,
                              hipStream_t stream) {
    const float* x      = (const float*)d_in[0];  // (S,B,D) f32
    const float* W      = (const float*)d_in[1];  // (3,D,D)
    const float* bias   = (const float*)d_in[2];  // (3,D)
    const float* alphas = (const float*)d_in[3];  // (3,)
    const float* betas  = (const float*)d_in[4];  // (3,)
    float*       out    = (float*)d_out;          // (S,B,D)
    float*       tfm    = (float*)d_ws;           // 3 * 8192 * 512 f32 = 48 MB

    // 3 planes * (8192/64 M-tiles) * (512/16 N-tiles) = 12288 waves / 8 per block
    gemm_wmma_f32<<<1536, 256, 0, stream>>>(x, W, bias, tfm);
    // B * S/8 = 8 * 128 = 1024 blocks
    window_fuse<<<1024, 256, 0, stream>>>(x, tfm, alphas, betas, out);
}